// VecSmoothAP_12335146074338
// MI455X (gfx1250) — compile-verified
//
#include <hip/hip_runtime.h>
#include <hip/hip_bf16.h>

#define L_DIM 16
#define P_DIM 768
#define D_DIM 256
#define N_TOT (L_DIM * P_DIM)      // 12288, divisible by 256

// sigmoid((sj-si)/0.01) = 1/(1+exp2((si-sj)*100*log2(e)))
#define SIG_SCALE 144.26950408889634f

typedef float v2f __attribute__((ext_vector_type(2)));
typedef float v8f __attribute__((ext_vector_type(8)));

// ---------------------------------------------------------------------------
// Phase 1: sims[16][768] = landmarks @ patches^T via V_WMMA_F32_16X16X4_F32.
// One wave (32 threads) per 16-wide column tile; 64 WMMAs over K=256.
// A 16x4 f32 layout: lanes 0-15 -> M=lane, Khalf 0; lanes 16-31 -> Khalf 1.
// B 4x16 mirrors A with N on lanes. Each lane loads a contiguous K pair (b64).
// ---------------------------------------------------------------------------
__global__ void sims_gemm_wmma(const float* __restrict__ lm,
                               const float* __restrict__ pf,
                               float* __restrict__ sims) {
  const int tile = blockIdx.x;        // 0..47
  const int lane = threadIdx.x;       // wave32
  const int kh   = lane >> 4;         // which K half-pair this lane holds
  const int mr   = lane & 15;         // M row (A) / N col (B)
  const float* arow = lm + mr * D_DIM;
  const float* brow = pf + (tile * 16 + mr) * D_DIM;

  v8f acc = {};
  #pragma unroll 8
  for (int kc = 0; kc < D_DIM / 4; ++kc) {
    const int kb = kc * 4 + kh * 2;
    v2f a = *(const v2f*)(arow + kb);
    v2f b = *(const v2f*)(brow + kb);
    acc = __builtin_amdgcn_wmma_f32_16x16x4_f32(
        /*neg_a=*/false, a, /*neg_b=*/false, b,
        /*c_mod=*/(short)0, acc, /*reuse_a=*/false, /*reuse_b=*/false);
  }

  // C/D layout: VGPR v -> M = v + 8*lanehalf, N = lane&15
  const int n = tile * 16 + mr;
  #pragma unroll
  for (int v = 0; v < 8; ++v)
    sims[(kh * 8 + v) * P_DIM + n] = acc[v];
}

// ---------------------------------------------------------------------------
// Phase 2: deterministic order-preserving compaction (single block, 8 waves).
//   cols[k]    = {sims[j], pos[j]?1:0}  for j where pn[j]   (n_pn entries)
//   rowSims[r] = sims[i]                for i where pos[i]  (n_pos entries)
// ---------------------------------------------------------------------------
__global__ void compact_kernel(const float* __restrict__ sims,
                               const int* __restrict__ pos,
                               const int* __restrict__ pn,
                               float2* __restrict__ cols,
                               float* __restrict__ rowSims,
                               int* __restrict__ counts) {
  __shared__ int wcnt_pn[8], wcnt_pos[8];
  __shared__ int wbase_pn[8], wbase_pos[8];
  __shared__ int run_pn, run_pos;
  const int tid  = threadIdx.x;
  const int w    = tid >> 5;          // wave32 index within block
  const int lane = tid & 31;
  if (tid == 0) { run_pn = 0; run_pos = 0; }
  __syncthreads();

  for (int base = 0; base < N_TOT; base += 256) {
    const int idx   = base + tid;
    const int pnv   = pn[idx];
    const int posv  = pos[idx];
    const float sv  = (pnv | posv) ? sims[idx] : 0.0f;
    const unsigned long long mpn   = __ballot(pnv != 0);
    const unsigned long long mpos  = __ballot(posv != 0);
    const unsigned long long lmask = (1ull << lane) - 1ull;
    if (lane == 0) { wcnt_pn[w] = (int)__popcll(mpn); wcnt_pos[w] = (int)__popcll(mpos); }
    __syncthreads();
    if (tid == 0) {
      int a = run_pn, b = run_pos;
      for (int i = 0; i < 8; ++i) {
        wbase_pn[i]  = a; a += wcnt_pn[i];
        wbase_pos[i] = b; b += wcnt_pos[i];
      }
      run_pn = a; run_pos = b;
    }
    __syncthreads();
    if (pnv)  cols[wbase_pn[w]  + (int)__popcll(mpn  & lmask)] =
                  make_float2(sv, posv ? 1.0f : 0.0f);
    if (posv) rowSims[wbase_pos[w] + (int)__popcll(mpos & lmask)] = sv;
    __syncthreads();
  }
  if (tid == 0) { counts[0] = run_pn; counts[1] = run_pos; }
}

// ---------------------------------------------------------------------------
// Phase 3: per positive row i: contrib = (1+Σ sig*pos_j)/(1+Σ sig) over pn cols.
// Grid-stride over rows; fixed-order shared-mem tree reduction (deterministic).
// ---------------------------------------------------------------------------
__global__ void rank_kernel(const float2* __restrict__ cols,
                            const float* __restrict__ rowSims,
                            const int* __restrict__ counts,
                            float* __restrict__ contrib) {
  __shared__ float red_a[256], red_b[256];
  const int n_pn  = counts[0];
  const int n_pos = counts[1];
  for (int r = blockIdx.x; r < n_pos; r += gridDim.x) {
    const float si = rowSims[r];
    float s_pn = 0.0f, s_pos = 0.0f;
    for (int j = threadIdx.x; j < n_pn; j += blockDim.x) {
      const float2 c  = cols[j];
      const float  x  = (si - c.x) * SIG_SCALE;          // = -(sj-si)/T * log2e
      const float  e  = exp2f(x);                         // v_exp_f32
      const float  sg = __builtin_amdgcn_rcpf(1.0f + e);  // v_rcp_f32
      s_pn  += sg;
      s_pos += sg * c.y;
    }
    red_a[threadIdx.x] = s_pn;
    red_b[threadIdx.x] = s_pos;
    __syncthreads();
    for (int off = 128; off > 0; off >>= 1) {
      if (threadIdx.x < off) {
        red_a[threadIdx.x] += red_a[threadIdx.x + off];
        red_b[threadIdx.x] += red_b[threadIdx.x + off];
      }
      __syncthreads();
    }
    if (threadIdx.x == 0)
      contrib[r] = (1.0f + red_b[0]) / (1.0f + red_a[0]);
    __syncthreads();
  }
}

// ---------------------------------------------------------------------------
// Phase 4: out = -Σ contrib / n_pos (single block, fixed-order reduction).
// ---------------------------------------------------------------------------
__global__ void final_kernel(const float* __restrict__ contrib,
                             const int* __restrict__ counts,
                             float* __restrict__ out) {
  __shared__ float red[256];
  const int n_pos = counts[1];
  float s = 0.0f;
  for (int i = threadIdx.x; i < n_pos; i += 256) s += contrib[i];
  red[threadIdx.x] = s;
  __syncthreads();
  for (int off = 128; off > 0; off >>= 1) {
    if (threadIdx.x < off) red[threadIdx.x] += red[threadIdx.x + off];
    __syncthreads();
  }
  if (threadIdx.x == 0)
    out[0] = (n_pos > 0) ? (-red[0] / (float)n_pos) : 0.0f;
}

extern "C" void kernel_launch(void* const* d_in, const int* in_sizes, int n_in,
                              void* d_out, int out_size, void* d_ws, size_t ws_size,
                              hipStream_t stream) {
  const float* lm  = (const float*)d_in[0];  // [16, 256]
  const float* pf  = (const float*)d_in[1];  // [768, 256]
  const int*   pos = (const int*)d_in[2];    // [12288]
  const int*   pn  = (const int*)d_in[3];    // [12288]

  char* ws = (char*)d_ws;
  float*  sims    = (float*)(ws);                 //  48 KB: N f32
  float2* cols    = (float2*)(ws + 48 * 1024);    //  96 KB: N float2 max
  float*  rowSims = (float*)(ws + 144 * 1024);    //  48 KB
  float*  contrib = (float*)(ws + 192 * 1024);    //  48 KB
  int*    counts  = (int*)(ws + 240 * 1024);      //  8 B

  sims_gemm_wmma<<<48, 32, 0, stream>>>(lm, pf, sims);
  compact_kernel<<<1, 256, 0, stream>>>(sims, pos, pn, cols, rowSims, counts);
  rank_kernel<<<256, 256, 0, stream>>>(cols, rowSims, counts, contrib);
  final_kernel<<<1, 256, 0, stream>>>(contrib, counts, (float*)d_out);
}